// CrossAggregator_73847667687537
// MI455X (gfx1250) — compile-verified
//
#include <hip/hip_runtime.h>
#include <math.h>

#define N_ROWS 10000
#define S_NEI  20
#define D_IN   512
#define D_OUT  512
#define ATTEN  64
#define K_TOT  1024

// GEMM tiling
#define TM 128            // rows per block (8 waves x 16)
#define TN 64             // output cols per block
#define KC 32             // K chunk staged in LDS (double buffered)
#define NC (K_TOT / KC)   // 32 chunks; chunks 0..15 self half, 16..31 neigh half
#define SA 36             // LDS row stride (floats): 32 k + 4 pad
                          //  -> 144B rows: 16B-aligned for async b128, 8B-aligned b64 frags,
                          //     lane*36 mod 64 is a permutation -> conflict-free frag reads

typedef float v2f __attribute__((ext_vector_type(2)));
typedef float v4f __attribute__((ext_vector_type(4)));
typedef float v8f __attribute__((ext_vector_type(8)));

// ---------------------------------------------------------------------------
// Kernel 0: collapse the attention head.
// sv[d]  = sum_a self_atten[d][a]  * v[a]
// nvv[d] = sum_a neigh_atten[d][a] * v[a]
// ---------------------------------------------------------------------------
__global__ void atten_vec_kernel(const float* __restrict__ self_atten,
                                 const float* __restrict__ neigh_atten,
                                 const float* __restrict__ v,
                                 float* __restrict__ sv,
                                 float* __restrict__ nvv) {
    __shared__ float vs[ATTEN];
    int t = threadIdx.x;
    if (t < ATTEN) vs[t] = v[t];
    __syncthreads();
    for (int d = t; d < D_IN; d += blockDim.x) {
        const float* sa = self_atten  + (size_t)d * ATTEN;
        const float* na = neigh_atten + (size_t)d * ATTEN;
        float a0 = 0.f, a1 = 0.f;
        #pragma unroll 8
        for (int a = 0; a < ATTEN; ++a) { a0 += sa[a] * vs[a]; a1 += na[a] * vs[a]; }
        sv[d] = a0;
        nvv[d] = a1;
    }
}

// ---------------------------------------------------------------------------
// Kernel 0b: transpose stacked weights into Wt[col][k], k = 0..1023
// (k < 512 -> self_weights, else neigh_weights). 2 MB once; makes B chunks
// contiguous along K in global memory (async-copy + ds_load_b64 friendly).
// ---------------------------------------------------------------------------
__global__ void __launch_bounds__(256)
wtrans_kernel(const float* __restrict__ sw,
              const float* __restrict__ nw,
              float* __restrict__ Wt) {
    __shared__ float tile[32][33];
    int kb = blockIdx.x * 32;              // k block   (grid.x = 32)
    int cb = blockIdx.y * 32;              // col block (grid.y = 16)
    int tx = threadIdx.x & 31;
    int ty = threadIdx.x >> 5;             // 32 x 8 threads
    #pragma unroll
    for (int i = 0; i < 32; i += 8) {
        int k = kb + ty + i;
        const float* W = (k < D_IN) ? (sw + (size_t)k * D_OUT)
                                    : (nw + (size_t)(k - D_IN) * D_OUT);
        tile[ty + i][tx] = W[cb + tx];
    }
    __syncthreads();
    #pragma unroll
    for (int i = 0; i < 32; i += 8) {
        int col = cb + ty + i;
        Wt[(size_t)col * K_TOT + kb + tx] = tile[tx][ty + i];
    }
}

// ---------------------------------------------------------------------------
// Kernel 1: per-row (one block per n): neighbor softmax + weighted sum,
// then the collapsed attention gate scalars wS[n], wN[n].
// Streams 410 MB of neigh_vecs -> memory bound (~18us at 23.3 TB/s).
// ---------------------------------------------------------------------------
__global__ void __launch_bounds__(256)
rowprep_kernel(const float* __restrict__ self_vecs,
               const float* __restrict__ neigh_vecs,
               const float* __restrict__ neigh_weight,
               const int*   __restrict__ neigh_column,
               const float* __restrict__ alpha,
               const float* __restrict__ sv,
               const float* __restrict__ nvv,
               float* __restrict__ neigh_sum,
               float* __restrict__ wS,
               float* __restrict__ wN) {
    int n = blockIdx.x;
    int t = threadIdx.x;

    __shared__ float es[S_NEI];
    __shared__ float coef[S_NEI];
    __shared__ float ssum;
    __shared__ float redp[256];
    __shared__ float redq[256];

    if (t < S_NEI) {
        int col = neigh_column[n * S_NEI + t];
        es[t] = expf(alpha[col]);
    }
    __syncthreads();
    if (t == 0) {
        float s = 0.f;
        #pragma unroll
        for (int i = 0; i < S_NEI; ++i) s += es[i];
        ssum = s;
    }
    __syncthreads();
    if (t < S_NEI) coef[t] = neigh_weight[n * S_NEI + t] * es[t] / ssum;
    __syncthreads();

    int d = 2 * t;
    const float* base = neigh_vecs + (size_t)n * S_NEI * D_IN + d;
    float acc0 = 0.f, acc1 = 0.f;
    #pragma unroll
    for (int s = 0; s < S_NEI; ++s) {
        v2f x = *(const v2f*)(base + (size_t)s * D_IN);
        float c = coef[s];
        acc0 += c * x.x;
        acc1 += c * x.y;
    }
    v2f outv; outv.x = acc0; outv.y = acc1;
    *(v2f*)(neigh_sum + (size_t)n * D_IN + d) = outv;

    v2f sx = *(const v2f*)(self_vecs + (size_t)n * D_IN + d);
    redp[t] = sx.x * sv[d] + sx.y * sv[d + 1];
    redq[t] = acc0 * nvv[d] + acc1 * nvv[d + 1];
    __syncthreads();
    #pragma unroll
    for (int off = 128; off > 0; off >>= 1) {
        if (t < off) { redp[t] += redp[t + off]; redq[t] += redq[t + off]; }
        __syncthreads();
    }
    if (t == 0) {
        float p = redp[0], q = redq[0];
        float a_self  = expf(tanhf(2.f * p));
        float a_neigh = expf(tanhf(q + p));
        float inv = 1.f / (a_self + a_neigh);
        wS[n] = a_self * inv;
        wN[n] = a_neigh * inv;
    }
}

// ---------------------------------------------------------------------------
// Kernel 2: fused GEMM, K = 1024 concatenation:
//   out[n,j] = relu( wS[n]*(self_vecs@self_w)[n,j] + wN[n]*(neigh_sum@neigh_w)[n,j] )
// V_WMMA_F32_16X16X4_F32 compute; global->LDS via double-buffered
// global_load_async_to_lds_b128 (ASYNCcnt), both A and B frags are single
// ds_load_b64 (no repacking movs). Row gate folded into the A fragment.
// ---------------------------------------------------------------------------
__global__ void __launch_bounds__(256)
fused_gemm_kernel(const float* __restrict__ self_vecs,
                  const float* __restrict__ neigh_sum,
                  const float* __restrict__ Wt,
                  const float* __restrict__ wS,
                  const float* __restrict__ wN,
                  float* __restrict__ out) {
    __shared__ float As[2][TM * SA];   // 2 * 128*36*4 = 36864 B
    __shared__ float Bs[2][TN * SA];   // 2 *  64*36*4 = 18432 B

    const int t    = threadIdx.x;
    const int wave = t >> 5;
    const int lane = t & 31;
    const int half = lane >> 4;        // 0: lanes 0-15, 1: lanes 16-31
    const int lm   = lane & 15;
    const int m_block   = blockIdx.x * TM;
    const int col_block = blockIdx.y * TN;

    // per-lane row gate (A-frag row is lane&15 for both halves)
    int arow = m_block + wave * 16 + lm;
    if (arow > N_ROWS - 1) arow = N_ROWS - 1;
    const float w_self  = wS[arow];
    const float w_neigh = wN[arow];

    // async stage one K-chunk into LDS buffer `buf`
    auto issue = [&](int ci, int buf) {
        const int k0 = ci * KC;
        const bool isSelf = (k0 < D_IN);
        const float* Asrc = isSelf ? self_vecs : neigh_sum;
        const int kOff = isSelf ? k0 : (k0 - D_IN);
        // A chunk: 128 rows x 32 k = 1024 x 16B; 4 ops/thread
        #pragma unroll
        for (int i = 0; i < 4; ++i) {
            int lin = t + i * 256;
            int row = lin >> 3;                // 8 float4 per row
            int kq  = lin & 7;
            int rg  = m_block + row;
            if (rg > N_ROWS - 1) rg = N_ROWS - 1;   // clamp ragged edge
            const float* g = Asrc + (size_t)rg * D_IN + kOff + kq * 4;
            unsigned l = (unsigned)(uintptr_t)&As[buf][row * SA + kq * 4];
            asm volatile("global_load_async_to_lds_b128 %0, %1, off"
                         :: "v"(l), "v"(g) : "memory");
        }
        // B chunk: 64 cols x 32 k = 512 x 16B; 2 ops/thread (Wt is K-contig)
        #pragma unroll
        for (int i = 0; i < 2; ++i) {
            int lin = t + i * 256;
            int col = lin >> 3;
            int kq  = lin & 7;
            const float* g = Wt + (size_t)(col_block + col) * K_TOT + k0 + kq * 4;
            unsigned l = (unsigned)(uintptr_t)&Bs[buf][col * SA + kq * 4];
            asm volatile("global_load_async_to_lds_b128 %0, %1, off"
                         :: "v"(l), "v"(g) : "memory");
        }
    };

    v8f c[4];
    #pragma unroll
    for (int j = 0; j < 4; ++j) c[j] = (v8f){0.f,0.f,0.f,0.f,0.f,0.f,0.f,0.f};

    issue(0, 0);
    for (int ci = 0; ci < NC; ++ci) {
        const int buf = ci & 1;
        if (ci + 1 < NC) {
            issue(ci + 1, buf ^ 1);                         // prefetch next chunk
            asm volatile("s_wait_asynccnt 0x6" ::: "memory"); // chunk ci landed (6 newer outstanding)
        } else {
            asm volatile("s_wait_asynccnt 0x0" ::: "memory");
        }
        __syncthreads();

        const float wrow = (ci * KC < D_IN) ? w_self : w_neigh;
        const float* ap = &As[buf][(wave * 16 + lm) * SA];
        const float* bp = &Bs[buf][0];
        #pragma unroll
        for (int kk = 0; kk < KC; kk += 4) {
            // A frag 16x4: lanes 0-15 -> K = kk,kk+1 ; lanes 16-31 -> K = kk+2,kk+3
            v2f a = *(const v2f*)(ap + kk + 2 * half);
            a *= wrow;                                     // fold per-row gate
            #pragma unroll
            for (int j = 0; j < 4; ++j) {
                // B frag (K-major LDS): contiguous pair -> single ds_load_b64
                v2f b = *(const v2f*)(bp + (j * 16 + lm) * SA + kk + 2 * half);
                c[j] = __builtin_amdgcn_wmma_f32_16x16x4_f32(
                           false, a, false, b, (short)0, c[j], false, false);
            }
        }
        __syncthreads();
    }

    // Store with ReLU; C/D layout: vgpr i holds row (i + 8*half), col = lane%16.
    const int wave_row0 = m_block + wave * 16;
    #pragma unroll
    for (int j = 0; j < 4; ++j) {
        int coln = col_block + j * 16 + lm;
        #pragma unroll
        for (int i = 0; i < 8; ++i) {
            int row = wave_row0 + i + 8 * half;
            if (row < N_ROWS) {
                out[(size_t)row * D_OUT + coln] = fmaxf(c[j][i], 0.f);
            }
        }
    }
}

// ---------------------------------------------------------------------------
extern "C" void kernel_launch(void* const* d_in, const int* in_sizes, int n_in,
                              void* d_out, int out_size, void* d_ws, size_t ws_size,
                              hipStream_t stream) {
    const float* self_vecs     = (const float*)d_in[0];
    const float* neigh_vecs    = (const float*)d_in[1];
    const float* neigh_weight  = (const float*)d_in[2];
    const int*   neigh_column  = (const int*)  d_in[3];
    const float* neigh_weights = (const float*)d_in[4];
    const float* self_weights  = (const float*)d_in[5];
    const float* alpha         = (const float*)d_in[6];
    const float* self_atten    = (const float*)d_in[7];
    const float* neigh_atten   = (const float*)d_in[8];
    const float* v             = (const float*)d_in[9];
    float* out = (float*)d_out;

    // Workspace (floats): neigh_sum [N*D] | wS [N] | wN [N] | sv [D] | nvv [D] | Wt [512*1024]
    float* ws        = (float*)d_ws;
    float* neigh_sum = ws;
    float* wS        = neigh_sum + (size_t)N_ROWS * D_IN;
    float* wN        = wS + N_ROWS;
    float* sv        = wN + N_ROWS;
    float* nvv       = sv + D_IN;
    float* Wt        = nvv + D_IN;

    atten_vec_kernel<<<1, 256, 0, stream>>>(self_atten, neigh_atten, v, sv, nvv);

    wtrans_kernel<<<dim3(K_TOT / 32, D_OUT / 32), 256, 0, stream>>>(self_weights,
                                                                    neigh_weights, Wt);

    rowprep_kernel<<<N_ROWS, 256, 0, stream>>>(self_vecs, neigh_vecs, neigh_weight,
                                               neigh_column, alpha, sv, nvv,
                                               neigh_sum, wS, wN);

    dim3 grid((N_ROWS + TM - 1) / TM, D_OUT / TN);
    fused_gemm_kernel<<<grid, 256, 0, stream>>>(self_vecs, neigh_sum, Wt,
                                                wS, wN, out);
}